// GCN_84456236908865
// MI455X (gfx1250) — compile-verified
//
#include <hip/hip_runtime.h>
#include <hip/hip_bf16.h>

// ---------------------------------------------------------------------------
// GCN forward on gfx1250 (MI455X).
//   in-degree count -> dinv -> CSR-by-dst (scan + fill + segment sort)
//   -> [WMMA-f32 GEMM (async-to-LDS staging) -> fused CSR gather+BN+ReLU] x2
//   -> deterministic per-graph pooling -> head GEMM.
// No float atomics anywhere: aggregation is a sorted gather, pooling is a
// fixed reduction tree. h/hw (25.6MB each) are L2-resident (192MB L2).
// ---------------------------------------------------------------------------

typedef __attribute__((ext_vector_type(2))) float v2f;
typedef __attribute__((ext_vector_type(8))) float v8f;

// --- zero int buffer --------------------------------------------------------
__global__ void zero_int_kernel(int* __restrict__ p, int n) {
    int i = blockIdx.x * blockDim.x + threadIdx.x;
    if (i < n) p[i] = 0;
}

// --- in-degree count over dst (int atomics: exact & deterministic) ----------
__global__ void cnt_kernel(const int* __restrict__ dstIdx, int* __restrict__ cnt, int nedges) {
    int e = blockIdx.x * blockDim.x + threadIdx.x;
    if (e >= nedges) return;
    atomicAdd(&cnt[dstIdx[e]], 1);
}

// --- dinv = rsqrt(indeg + 1) ------------------------------------------------
__global__ void dinv_kernel(const int* __restrict__ cnt, float* __restrict__ dinv, int n) {
    int i = blockIdx.x * blockDim.x + threadIdx.x;
    if (i < n) dinv[i] = rsqrtf((float)cnt[i] + 1.0f);
}

// --- single-block exclusive scan of cnt[0..n) -> offs[0..n] -----------------
__global__ __launch_bounds__(1024) void scan_kernel(const int* __restrict__ cnt,
                                                    int* __restrict__ offs, int n) {
    __shared__ int part[1024];
    const int tid = threadIdx.x;
    const int chunk = (n + 1023) / 1024;
    const int a = tid * chunk;
    const int b = (a + chunk < n) ? a + chunk : n;
    int s = 0;
    for (int i = a; i < b; ++i) s += cnt[i];
    part[tid] = s;
    __syncthreads();
    for (int d = 1; d < 1024; d <<= 1) {
        int v = 0;
        if (tid >= d) v = part[tid - d];
        __syncthreads();
        if (tid >= d) part[tid] += v;
        __syncthreads();
    }
    int run = (tid == 0) ? 0 : part[tid - 1];
    for (int i = a; i < b; ++i) { offs[i] = run; run += cnt[i]; }
    if (tid == 0) offs[n] = part[1023];
}

// --- CSR fill: csr[offs[dst]+cursor++] = src --------------------------------
__global__ void fill_kernel(const int* __restrict__ srcIdx, const int* __restrict__ dstIdx,
                            const int* __restrict__ offs, int* __restrict__ cursor,
                            int* __restrict__ csr, int nedges) {
    int e = blockIdx.x * blockDim.x + threadIdx.x;
    if (e >= nedges) return;
    int d = dstIdx[e];
    int pos = offs[d] + atomicAdd(&cursor[d], 1);
    csr[pos] = srcIdx[e];
}

// --- per-node insertion sort of CSR segment (restores determinism) ----------
__global__ void sort_kernel(const int* __restrict__ offs, int* __restrict__ csr, int n) {
    int node = blockIdx.x * blockDim.x + threadIdx.x;
    if (node >= n) return;
    int a = offs[node], b = offs[node + 1];
    for (int i = a + 1; i < b; ++i) {
        int key = csr[i];
        int j = i - 1;
        while (j >= a && csr[j] > key) { csr[j + 1] = csr[j]; --j; }
        csr[j + 1] = key;
    }
}

// --- graph boundaries from sorted batch: graph g = nodes [bound[g],bound[g+1])
__global__ void bounds_kernel(const int* __restrict__ batch, int* __restrict__ bound,
                              int n, int G) {
    int i = blockIdx.x * blockDim.x + threadIdx.x;
    if (i >= n) return;
    if (i == 0) {
        for (int g = 0; g <= batch[0]; ++g) bound[g] = 0;
        for (int g = batch[n - 1] + 1; g <= G; ++g) bound[g] = n;
    } else {
        int p = batch[i - 1], c = batch[i];
        for (int g = p + 1; g <= c; ++g) bound[g] = i;
    }
}

// --- H = X @ W  (N x 64) @ (64 x 64), f32 WMMA ------------------------------
// Block: 256 threads = 8 waves; each wave computes a 16x64 output tile.
// LDS staging via global_load_async_to_lds_b128 (ASYNCcnt) for full blocks;
// guarded scalar fallback for the tail block. Rows padded to 65 floats.
__global__ __launch_bounds__(256) void gemm64_wmma_kernel(
        const float* __restrict__ X, const float* __restrict__ W,
        float* __restrict__ H, int nrows) {
    __shared__ float sW[64 * 65];
    __shared__ float sX[128 * 65];

    const int tid = threadIdx.x;
    const int rowBase = blockIdx.x * 128;

    // stage W (64x64 [k][n]): 1024 b128 chunks, async copy to LDS
    for (int i4 = tid; i4 < 1024; i4 += 256) {
        int k = i4 >> 4, c4 = (i4 & 15) << 2;
        unsigned lds = (unsigned)(size_t)(&sW[k * 65 + c4]);
        const float* g = W + k * 64 + c4;
        asm volatile("global_load_async_to_lds_b128 %0, %1, off"
                     :: "v"(lds), "v"(g) : "memory");
    }
    if (rowBase + 128 <= nrows) {
        // full block: async copy 128x64 X tile (2048 b128 chunks)
        for (int i4 = tid; i4 < 2048; i4 += 256) {
            int r = i4 >> 4, c4 = (i4 & 15) << 2;
            unsigned lds = (unsigned)(size_t)(&sX[r * 65 + c4]);
            const float* g = X + (long long)(rowBase + r) * 64 + c4;
            asm volatile("global_load_async_to_lds_b128 %0, %1, off"
                         :: "v"(lds), "v"(g) : "memory");
        }
    } else {
        // tail block: guarded scalar path with zero fill
        for (int idx = tid; idx < 128 * 64; idx += 256) {
            int r = idx >> 6, c = idx & 63;
            int grow = rowBase + r;
            sX[r * 65 + c] = (grow < nrows) ? X[(long long)grow * 64 + c] : 0.0f;
        }
    }
    asm volatile("s_wait_asynccnt 0" ::: "memory");
    __syncthreads();

    // WMMA f32 16x16x4 fragment layouts (ISA 7.12.2):
    //   A: lanes 0-15 {v0,v1}=A[M=lane,K=0..1]; lanes 16-31 K=2..3
    //   B: lanes 0-15 {v0,v1}=B[K=0..1,N=lane]; lanes 16-31 K=2..3
    //   C/D: VGPR r -> (M=r, N=lane) / (M=r+8, N=lane-16)
    const int wv = tid >> 5;
    const int lane = tid & 31;
    const int hi = lane >> 4;
    const int m = lane & 15;

    const float* xrow = &sX[(wv * 16 + m) * 65];

    v8f acc[4] = {};
#pragma unroll
    for (int kk = 0; kk < 16; ++kk) {
        const int kb = kk * 4 + 2 * hi;
        v2f a;
        a.x = xrow[kb];
        a.y = xrow[kb + 1];
#pragma unroll
        for (int j = 0; j < 4; ++j) {
            v2f b;
            b.x = sW[kb * 65 + j * 16 + m];
            b.y = sW[(kb + 1) * 65 + j * 16 + m];
            acc[j] = __builtin_amdgcn_wmma_f32_16x16x4_f32(
                false, a, false, b, (short)0, acc[j], false, false);
        }
    }

    const int rowTile = rowBase + wv * 16;
#pragma unroll
    for (int j = 0; j < 4; ++j) {
#pragma unroll
        for (int r = 0; r < 8; ++r) {
            int row = rowTile + r + 8 * hi;
            if (row < nrows) H[(long long)row * 64 + j * 16 + m] = acc[j][r];
        }
    }
}

// --- fused gather + self + bias + BN + ReLU ---------------------------------
// out[n] = relu(bn( dinv[n]*sum_{s in N(n)} hw[s]*dinv[s] + hw[n]*dinv[n]^2 + b ))
// 16 threads per node, float4 per thread; sorted CSR -> deterministic sums.
__global__ void gather_bn_kernel(const int* __restrict__ offs, const int* __restrict__ csr,
                                 const float* __restrict__ dinv, const float* __restrict__ hw,
                                 const float* __restrict__ bias,
                                 const float* __restrict__ gamma, const float* __restrict__ beta,
                                 const float* __restrict__ mean, const float* __restrict__ var,
                                 float* __restrict__ out, int nnodes) {
    long long tid = (long long)blockIdx.x * blockDim.x + threadIdx.x;
    int node = (int)(tid >> 4);
    if (node >= nnodes) return;
    int f4 = (int)(tid & 15) * 4;

    int a = offs[node], b = offs[node + 1];
    float ax = 0.0f, ay = 0.0f, az = 0.0f, aw = 0.0f;
    for (int j = a; j < b; ++j) {
        int s = csr[j];
        float w = dinv[s];
        const float4 v = *reinterpret_cast<const float4*>(hw + (long long)s * 64 + f4);
        ax += v.x * w; ay += v.y * w; az += v.z * w; aw += v.w * w;
    }
    float dn = dinv[node];
    float sw = dn * dn;
    const float4 self = *reinterpret_cast<const float4*>(hw + (long long)node * 64 + f4);
    const float4 bz = *reinterpret_cast<const float4*>(bias + f4);
    const float4 gm = *reinterpret_cast<const float4*>(gamma + f4);
    const float4 bt = *reinterpret_cast<const float4*>(beta + f4);
    const float4 mn = *reinterpret_cast<const float4*>(mean + f4);
    const float4 vr = *reinterpret_cast<const float4*>(var + f4);

    float vx = ax * dn + self.x * sw + bz.x;
    float vy = ay * dn + self.y * sw + bz.y;
    float vz = az * dn + self.z * sw + bz.z;
    float vw = aw * dn + self.w * sw + bz.w;

    vx = (vx - mn.x) * rsqrtf(vr.x + 1e-5f) * gm.x + bt.x;
    vy = (vy - mn.y) * rsqrtf(vr.y + 1e-5f) * gm.y + bt.y;
    vz = (vz - mn.z) * rsqrtf(vr.z + 1e-5f) * gm.z + bt.z;
    vw = (vw - mn.w) * rsqrtf(vr.w + 1e-5f) * gm.w + bt.w;

    float4 o;
    o.x = vx > 0.0f ? vx : 0.0f;
    o.y = vy > 0.0f ? vy : 0.0f;
    o.z = vz > 0.0f ? vz : 0.0f;
    o.w = vw > 0.0f ? vw : 0.0f;
    *reinterpret_cast<float4*>(out + (long long)node * 64 + f4) = o;
}

// --- deterministic pooling: one block per graph, fixed 4-way reduction ------
__global__ __launch_bounds__(256) void pool_graph_kernel(const float* __restrict__ h,
                                                         const int* __restrict__ bound,
                                                         float* __restrict__ pooled) {
    __shared__ float red[256];
    const int g = blockIdx.x;
    const int f = threadIdx.x & 63;
    const int q = threadIdx.x >> 6;
    const int s = bound[g], e = bound[g + 1];
    const int len = e - s;
    const int per = (len + 3) >> 2;
    int a = s + q * per;
    int b = a + per; if (b > e) b = e; if (a > e) a = e;
    float acc = 0.0f;
    for (int i = a; i < b; ++i) acc += h[(long long)i * 64 + f];
    red[threadIdx.x] = acc;
    __syncthreads();
    if (q == 0) pooled[g * 64 + f] = red[f] + red[64 + f] + red[128 + f] + red[192 + f];
}

// --- head: score[g][o] = sum_l pooled[l][g][:] @ headW[l][:,o] + headb[l][o]
__global__ void head_kernel(const float* __restrict__ pooled,
                            const float* __restrict__ headW, const float* __restrict__ headb,
                            float* __restrict__ out, int G) {
    int tid = blockIdx.x * blockDim.x + threadIdx.x;
    if (tid >= G * 16) return;
    int g = tid >> 4;
    int o = tid & 15;
    float s = 0.0f;
#pragma unroll
    for (int l = 0; l < 3; ++l) {
        const float* p = pooled + (l * G + g) * 64;
        const float* w = headW + l * 64 * 16 + o;
        float acc = 0.0f;
        for (int k = 0; k < 64; ++k) acc += p[k] * w[k * 16];
        s += acc + headb[l * 16 + o];
    }
    out[g * 16 + o] = s;
}

extern "C" void kernel_launch(void* const* d_in, const int* in_sizes, int n_in,
                              void* d_out, int out_size, void* d_ws, size_t ws_size,
                              hipStream_t stream) {
    (void)n_in; (void)ws_size;

    const float* x     = (const float*)d_in[0];
    const int*   eidx  = (const int*)d_in[1];   // [2, E]
    const int*   batch = (const int*)d_in[2];
    const float* convW = (const float*)d_in[3]; // [2][64][64]
    const float* convB = (const float*)d_in[4]; // [2][64]
    const float* bnG   = (const float*)d_in[5];
    const float* bnB   = (const float*)d_in[6];
    const float* bnM   = (const float*)d_in[7];
    const float* bnV   = (const float*)d_in[8];
    const float* headW = (const float*)d_in[9];  // [3][64][16]
    const float* headB = (const float*)d_in[10]; // [3][16]
    float* out = (float*)d_out;

    const int N = in_sizes[0] / 64;
    const int E = in_sizes[1] / 2;
    const int G = out_size / 16;
    const int* srcIdx = eidx;
    const int* dstIdx = eidx + E;

    // workspace layout
    char* ws = (char*)d_ws;
    size_t off = 0;
    auto take = [&](size_t bytes) -> char* {
        char* p = ws + off;
        off += (bytes + 255) & ~(size_t)255;
        return p;
    };
    float* dinv   = (float*)take((size_t)N * 4);
    int*   cnt    = (int*)  take((size_t)N * 4);
    int*   offs   = (int*)  take((size_t)(N + 1) * 4);
    int*   bound  = (int*)  take((size_t)(G + 1) * 4);
    int*   csr    = (int*)  take((size_t)E * 4);
    float* hw     = (float*)take((size_t)N * 64 * 4);
    float* h1     = (float*)take((size_t)N * 64 * 4);
    float* pooled = (float*)take((size_t)3 * G * 64 * 4);

    const int gN    = (N + 255) / 256;
    const int gE    = (E + 255) / 256;
    const int gGemm = (N + 127) / 128;
    const int gGath = (int)(((long long)N * 16 + 255) / 256);
    const int gHead = (G * 16 + 255) / 256;

    // ---- graph structure (CSR by dst) + normalization ----
    zero_int_kernel<<<gN, 256, 0, stream>>>(cnt, N);
    cnt_kernel<<<gE, 256, 0, stream>>>(dstIdx, cnt, E);
    dinv_kernel<<<gN, 256, 0, stream>>>(cnt, dinv, N);
    scan_kernel<<<1, 1024, 0, stream>>>(cnt, offs, N);
    zero_int_kernel<<<gN, 256, 0, stream>>>(cnt, N);
    fill_kernel<<<gE, 256, 0, stream>>>(srcIdx, dstIdx, offs, cnt, csr, E);
    sort_kernel<<<gN, 256, 0, stream>>>(offs, csr, N);
    bounds_kernel<<<gN, 256, 0, stream>>>(batch, bound, N, G);

    // ---- layer 0 ----
    gemm64_wmma_kernel<<<gGemm, 256, 0, stream>>>(x, convW, hw, N);
    gather_bn_kernel<<<gGath, 256, 0, stream>>>(offs, csr, dinv, hw, convB,
                                                bnG, bnB, bnM, bnV, h1, N);

    pool_graph_kernel<<<G, 256, 0, stream>>>(x,  bound, pooled);
    pool_graph_kernel<<<G, 256, 0, stream>>>(h1, bound, pooled + G * 64);

    // ---- layer 1 (h2 overwrites h1 after gemm consumed it) ----
    gemm64_wmma_kernel<<<gGemm, 256, 0, stream>>>(h1, convW + 64 * 64, hw, N);
    gather_bn_kernel<<<gGath, 256, 0, stream>>>(offs, csr, dinv, hw, convB + 64,
                                                bnG + 64, bnB + 64, bnM + 64, bnV + 64,
                                                h1, N);
    pool_graph_kernel<<<G, 256, 0, stream>>>(h1, bound, pooled + 2 * G * 64);

    // ---- head ----
    head_kernel<<<gHead, 256, 0, stream>>>(pooled, headW, headB, out, G);
}